// MeshDihedralAngleLoss_8117488189448
// MI455X (gfx1250) — compile-verified
//
#include <hip/hip_runtime.h>
#include <math.h>

// ---------------------------------------------------------------------------
// MeshDihedralAngleLoss for MI455X (gfx1250, wave32)
//   B=8, N=100000, E=300000.  Gather-bound: vertex data (19.2 MB) lives in the
//   192 MB L2; HBM traffic dominated by the streamed int32 index array.
//   Blocks are mapped per-batch so the inner loop is pure 32-bit math with
//   uniform (SGPR) batch base pointers; unconditional speculative prefetch.
//   Deterministic 2-stage reduction; wave sums use V_WMMA_F32_16X16X4_F32.
// ---------------------------------------------------------------------------

typedef __attribute__((ext_vector_type(2))) float v2f;
typedef __attribute__((ext_vector_type(8))) float v8f;

#define MB_    8
#define MN_    100000
#define ME_    300000
#define BPB_   256                    // blocks per batch
#define NBLK_  (MB_ * BPB_)           // 2048 stage-1 blocks -> 8 KB partials
#define THREADS_ 256
#define ESTRIDE_ (BPB_ * THREADS_)    // 65536 edges per sweep within a batch
#define CLIP_  (1.0f - 1e-7f)
#define PI_F_  3.14159265358979323846f

// Deterministic 32-lane wave sum via one V_WMMA_F32_16X16X4_F32.
// A (16x4 f32): lane L<16 -> A[L][0]=x; lane L>=16 -> A[L-16][2]=x; rest 0.
// B = all-ones (layout-independent). D[m][n] = x[m] + x[m+16].
// Sum of the 8 D VGPRs per lane = half-sum; shfl_xor(16) completes it.
// Requires EXEC == all ones: called unconditionally from full 256-thread blocks.
__device__ __forceinline__ float wave_sum32_wmma(float x) {
  v2f a;    a[0] = x;    a[1] = 0.0f;
  v2f ones; ones[0] = 1.0f; ones[1] = 1.0f;
  v8f c = {};
  c = __builtin_amdgcn_wmma_f32_16x16x4_f32(
      /*neg_a=*/false, a, /*neg_b=*/false, ones,
      /*c_mod=*/(short)0, c, /*reuse_a=*/false, /*reuse_b=*/false);
  float s = ((c[0] + c[1]) + (c[2] + c[3])) + ((c[4] + c[5]) + (c[6] + c[7]));
  s += __shfl_xor(s, 16, 32);
  return s;  // total of all 32 lanes, in every lane
}

__device__ __forceinline__ float block_reduce(float acc, float* lds,
                                              int tid) {
  const float s = wave_sum32_wmma(acc);
  const int lane = tid & 31;
  const int wv   = tid >> 5;
  if (lane == 0) lds[wv] = s;
  __syncthreads();
  float tot = 0.0f;
  if (tid == 0) {
    tot = lds[0];
#pragma unroll
    for (int w = 1; w < THREADS_ / 32; ++w) tot += lds[w];
  }
  return tot;  // valid in thread 0 only
}

// Dihedral angle per the reference:
//   na = cross(v[i2]-v[i0], v[i1]-v[i0]); nb = cross(v[i3]-v[i1], v[i0]-v[i1])
//   angle = pi - acos(clip(na_hat . nb_hat, -CLIP, CLIP))
__device__ __forceinline__ float dihedral_angle(const float* __restrict__ v,
                                                int i0, int i1, int i2, int i3) {
  const float p0x = v[3 * i0 + 0], p0y = v[3 * i0 + 1], p0z = v[3 * i0 + 2];
  const float p1x = v[3 * i1 + 0], p1y = v[3 * i1 + 1], p1z = v[3 * i1 + 2];
  const float p2x = v[3 * i2 + 0], p2y = v[3 * i2 + 1], p2z = v[3 * i2 + 2];
  const float p3x = v[3 * i3 + 0], p3y = v[3 * i3 + 1], p3z = v[3 * i3 + 2];

  // face 0: ea = p2 - p0, eb = p1 - p0
  const float ax = p2x - p0x, ay = p2y - p0y, az = p2z - p0z;
  const float bx = p1x - p0x, by = p1y - p0y, bz = p1z - p0z;
  const float nax = ay * bz - az * by;
  const float nay = az * bx - ax * bz;
  const float naz = ax * by - ay * bx;

  // face 1: ea = p3 - p1, eb = p0 - p1
  const float cx = p3x - p1x, cy = p3y - p1y, cz = p3z - p1z;
  const float dx = p0x - p1x, dy = p0y - p1y, dz = p0z - p1z;
  const float nbx = cy * dz - cz * dy;
  const float nby = cz * dx - cx * dz;
  const float nbz = cx * dy - cy * dx;

  const float la = fmaxf(sqrtf(nax * nax + nay * nay + naz * naz), 1e-12f);
  const float lb = fmaxf(sqrtf(nbx * nbx + nby * nby + nbz * nbz), 1e-12f);
  float dot = (nax * nbx + nay * nby + naz * nbz) / (la * lb);
  dot = fminf(fmaxf(dot, -CLIP_), CLIP_);
  return PI_F_ - acosf(dot);
}

__global__ __launch_bounds__(THREADS_) void mdal_partial_kernel(
    const float* __restrict__ vert1, const float* __restrict__ vert2,
    const int* __restrict__ ep, float* __restrict__ partials) {
  // Batch is uniform per block -> base pointers live in SGPRs.
  const int b   = (int)blockIdx.x / BPB_;
  const int blk = (int)blockIdx.x % BPB_;
  const float* __restrict__ v1  = vert1 + (size_t)b * (MN_ * 3);
  const float* __restrict__ v2  = vert2 + (size_t)b * (MN_ * 3);
  const int4*  __restrict__ q4  = (const int4*)ep + (size_t)b * ME_;

  float acc = 0.0f;
  for (int e = blk * THREADS_ + (int)threadIdx.x; e < ME_; e += ESTRIDE_) {
    // Speculative prefetch of next sweep's indices; invalid addresses are
    // silently dropped by hardware, so no guard branch needed.
    __builtin_prefetch(q4 + e + ESTRIDE_, 0, 0);  // global_prefetch_b8

    const int4 q = q4[e];                         // coalesced 16B index load
    const float a1 = dihedral_angle(v1, q.x, q.y, q.z, q.w);
    const float a2 = dihedral_angle(v2, q.x, q.y, q.z, q.w);
    const float d  = a1 - a2;
    acc = fmaf(d, d, acc);
  }

  __shared__ float lds[THREADS_ / 32];
  const float tot = block_reduce(acc, lds, (int)threadIdx.x);
  if (threadIdx.x == 0) partials[blockIdx.x] = tot;
}

__global__ __launch_bounds__(THREADS_) void mdal_final_kernel(
    const float* __restrict__ partials, float* __restrict__ out) {
  float acc = 0.0f;
  for (int i = (int)threadIdx.x; i < NBLK_; i += THREADS_)  // fixed order
    acc += partials[i];

  __shared__ float lds[THREADS_ / 32];
  const float tot = block_reduce(acc, lds, (int)threadIdx.x);
  if (threadIdx.x == 0) {
    // mean over batches of mean over edges == sum / (B*E)
    out[0] = tot * (float)(1.0 / ((double)MB_ * (double)ME_));
  }
}

extern "C" void kernel_launch(void* const* d_in, const int* in_sizes, int n_in,
                              void* d_out, int out_size, void* d_ws, size_t ws_size,
                              hipStream_t stream) {
  (void)in_sizes; (void)n_in; (void)out_size; (void)ws_size;
  const float* vert1 = (const float*)d_in[0];
  const float* vert2 = (const float*)d_in[1];
  const int*   ep    = (const int*)d_in[2];   // integer input -> int32 per harness
  float* partials = (float*)d_ws;             // NBLK_ floats = 8 KB scratch
  float* out      = (float*)d_out;

  mdal_partial_kernel<<<NBLK_, THREADS_, 0, stream>>>(vert1, vert2, ep, partials);
  mdal_final_kernel<<<1, THREADS_, 0, stream>>>(partials, out);
}